// MaskedAttention_12936441496276
// MI455X (gfx1250) — compile-verified
//
#include <hip/hip_runtime.h>
#include <math.h>
#include <stdint.h>

// ---------------- problem constants ----------------
#define B_    4
#define CIN   128
#define TQ    256
#define TS    257
#define TD    65
#define COUT  512
#define HID   256
#define NHEAD 8
#define EDIM  32
#define NPER  (TQ*TD)          // 16640, divisible by 64
#define GN_EPS 1e-5f

typedef float v2f __attribute__((ext_vector_type(2)));
typedef float v8f __attribute__((ext_vector_type(8)));
typedef unsigned int u32x4 __attribute__((ext_vector_type(4)));
typedef int i32x4 __attribute__((ext_vector_type(4)));
typedef int i32x8 __attribute__((ext_vector_type(8)));

static __device__ __forceinline__ v8f zero8() {
  v8f z = {0.f,0.f,0.f,0.f,0.f,0.f,0.f,0.f};
  return z;
}

// V_WMMA_F32_16X16X4_F32 : D(16x16 f32) = A(16x4 f32) * B(4x16 f32) + C
static __device__ __forceinline__ v8f wmma4(v2f a, v2f b, v8f c) {
  return __builtin_amdgcn_wmma_f32_16x16x4_f32(false, a, false, b, (short)0, c,
                                               false, false);
}

// ---------------- mask: bilinear 256x256 -> 16x16 ----------------
__global__ __launch_bounds__(256) void k_mask(const int* __restrict__ sm,
                                              float* __restrict__ mout) {
  const int b = blockIdx.x;
  const int t = threadIdx.x;           // 256 = 16x16
  const int ty = t >> 4, tx = t & 15;
  const float step = 255.0f / 15.0f;
  float cy = ty * step, cx = tx * step;
  int y0 = (int)floorf(cy); int y1 = min(y0 + 1, 255); float fy = cy - (float)y0;
  int x0 = (int)floorf(cx); int x1 = min(x0 + 1, 255); float fx = cx - (float)x0;
  const int* sb = sm + (size_t)b * 256 * 256;
  float v00 = (float)min(sb[y0*256 + x0], 1);
  float v01 = (float)min(sb[y0*256 + x1], 1);
  float v10 = (float)min(sb[y1*256 + x0], 1);
  float v11 = (float)min(sb[y1*256 + x1], 1);
  float v0 = v00 * (1.f - fy) + v10 * fy;
  float v1 = v01 * (1.f - fy) + v11 * fy;
  mout[b*256 + t] = v0 * (1.f - fx) + v1 * fx;
}

// ---------------- cst_pool: corr (B,CIN,TQ,257) -> corrP (B,CIN,TQ,65) ----------------
__global__ __launch_bounds__(256) void k_pool(const float* __restrict__ corr,
                                              float* __restrict__ dst,
                                              size_t total) {
  for (size_t idx = (size_t)blockIdx.x * blockDim.x + threadIdx.x; idx < total;
       idx += (size_t)gridDim.x * blockDim.x) {
    int d = (int)(idx % TD);
    size_t rest = idx / TD;
    int q = (int)(rest % TQ);
    rest /= TQ;
    int c = (int)(rest % CIN);
    int b = (int)(rest / CIN);
    const float* base = corr + (((size_t)b*CIN + c)*TQ + q)*TS;
    float v;
    if (d == 0) {
      v = base[0];
    } else {
      int p = d - 1;
      int j = p >> 3, i = p & 7;
      int r0 = 1 + (2*j)*16 + 2*i;
      v = 0.25f * (base[r0] + base[r0+1] + base[r0+16] + base[r0+17]);
    }
    dst[idx] = v;
  }
}

// ---------------- f32 WMMA GEMM ----------------
// Y[b,m,n] = W[m,:] * X'[b,:,n] (+bias); when HASCF, X' = relu(X*scale+shift).
// Each wave: two 16-row m-positions (mA, mA+128) x 64 n-cols -> 8 accumulators.
// Block covers 256 rows; grid = (NPER/64, M/256, B).
// Optional epilogue: per-block GN partial sums (group = 128-ch slab = row slab).
template <bool HASCF>
__global__ __launch_bounds__(256) void k_gemm_t(const float* __restrict__ W,
                                                const float* __restrict__ X,
                                                const float* __restrict__ bias,
                                                const float* __restrict__ xcoef,
                                                float* __restrict__ Y,
                                                float* __restrict__ part,
                                                int M, int K) {
  const int tid  = threadIdx.x;
  const int lane = tid & 31;
  const int wv   = tid >> 5;
  const int b    = blockIdx.z;
  const int mA   = blockIdx.y * 256 + wv * 16;
  const int mB   = mA + 128;
  const int n0   = blockIdx.x * 64;
  const int arow = lane & 15;
  const int kof  = (lane < 16) ? 0 : 2;
  const int ncol = lane & 15;

  const float* Xb  = X + (size_t)b * K * NPER;
  const float* wrA = W + (size_t)(mA + arow) * K;
  const float* wrB = W + (size_t)(mB + arow) * K;
  const float* cf  = HASCF ? (xcoef + (size_t)b * K * 2) : nullptr;

  v8f accA[4], accB[4];
#pragma unroll
  for (int t = 0; t < 4; ++t) { accA[t] = zero8(); accB[t] = zero8(); }

  for (int k4 = 0; k4 < K; k4 += 4) {
    v2f aA; aA.x = wrA[k4 + kof]; aA.y = wrA[k4 + kof + 1];
    v2f aB; aB.x = wrB[k4 + kof]; aB.y = wrB[k4 + kof + 1];
    const float* xr = Xb + (size_t)(k4 + kof) * NPER;
    float sc0 = 1.f, sh0 = 0.f, sc1 = 1.f, sh1 = 0.f;
    if (HASCF) {
      sc0 = cf[(k4+kof)*2];     sh0 = cf[(k4+kof)*2 + 1];
      sc1 = cf[(k4+kof+1)*2];   sh1 = cf[(k4+kof+1)*2 + 1];
    }
#pragma unroll
    for (int t = 0; t < 4; ++t) {
      int nn = n0 + t*16 + ncol;
      v2f bv;
      bv.x = xr[nn];
      bv.y = xr[nn + NPER];
      if (HASCF) {                    // fused GN-apply + relu on the input
        bv.x = fmaxf(bv.x * sc0 + sh0, 0.f);
        bv.y = fmaxf(bv.y * sc1 + sh1, 0.f);
      }
      accA[t] = wmma4(aA, bv, accA[t]);
      accB[t] = wmma4(aB, bv, accB[t]);
    }
  }

  float* Yb = Y + (size_t)b * M * NPER;
  const int rb = (lane < 16) ? 0 : 8;
  float sA = 0.f, s2A = 0.f, sB = 0.f, s2B = 0.f;
#pragma unroll
  for (int r = 0; r < 8; ++r) {
    const int ma = mA + rb + r;
    const int mb2 = mB + rb + r;
    const float bba = bias ? bias[ma]  : 0.f;
    const float bbb = bias ? bias[mb2] : 0.f;
#pragma unroll
    for (int t = 0; t < 4; ++t) {
      const int nn = n0 + t*16 + ncol;
      float vA = accA[t][r] + bba;
      float vB = accB[t][r] + bbb;
      Yb[(size_t)ma  * NPER + nn] = vA;
      Yb[(size_t)mb2 * NPER + nn] = vB;
      sA += vA; s2A += vA*vA;
      sB += vB; s2B += vB*vB;
    }
  }

  if (part) {   // deterministic per-block GN partials (fixed-order tree)
    __shared__ float r1[256], r2[256];
    r1[tid] = sA; r2[tid] = s2A;
    __syncthreads();
    for (int o = 128; o > 0; o >>= 1) {
      if (tid < o) { r1[tid] += r1[tid+o]; r2[tid] += r2[tid+o]; }
      __syncthreads();
    }
    if (tid == 0) {
      int bg = b*4 + blockIdx.y*2;
      part[((size_t)bg * gridDim.x + blockIdx.x)*2]     = r1[0];
      part[((size_t)bg * gridDim.x + blockIdx.x)*2 + 1] = r2[0];
    }
    __syncthreads();
    r1[tid] = sB; r2[tid] = s2B;
    __syncthreads();
    for (int o = 128; o > 0; o >>= 1) {
      if (tid < o) { r1[tid] += r1[tid+o]; r2[tid] += r2[tid+o]; }
      __syncthreads();
    }
    if (tid == 0) {
      int bg = b*4 + blockIdx.y*2 + 1;
      part[((size_t)bg * gridDim.x + blockIdx.x)*2]     = r1[0];
      part[((size_t)bg * gridDim.x + blockIdx.x)*2 + 1] = r2[0];
    }
  }
}

// ---------------- GN helpers (deterministic) ----------------
// reduce nparts partials per (b,grp) in fixed order. grid = 16 blocks
__global__ __launch_bounds__(256) void k_gn_reduceN(const float* __restrict__ part,
                                                    float* __restrict__ stats,
                                                    int nparts) {
  const int bg = blockIdx.x, tid = threadIdx.x;
  float s = 0.f, s2 = 0.f;
  for (int i = tid; i < nparts; i += 256) {
    s  += part[((size_t)bg*nparts + i)*2];
    s2 += part[((size_t)bg*nparts + i)*2 + 1];
  }
  __shared__ float r1[256], r2[256];
  r1[tid] = s; r2[tid] = s2;
  __syncthreads();
  for (int o = 128; o > 0; o >>= 1) {
    if (tid < o) { r1[tid] += r1[tid+o]; r2[tid] += r2[tid+o]; }
    __syncthreads();
  }
  if (tid == 0) { stats[2*bg] = r1[0]; stats[2*bg+1] = r2[0]; }
}

// per-(b,ch) affine coefficients: y = x*scale + shift
__global__ __launch_bounds__(256) void k_gn_coef(const float* __restrict__ stats,
                                                 const float* __restrict__ w,
                                                 const float* __restrict__ bb,
                                                 float* __restrict__ cf,
                                                 int C, float invc) {
  for (int i = blockIdx.x*blockDim.x + threadIdx.x; i < B_*C;
       i += gridDim.x*blockDim.x) {
    int b = i / C, ch = i % C;
    int grp = ch / (C/4);
    float s  = stats[2*(b*4+grp)];
    float s2 = stats[2*(b*4+grp)+1];
    float mu = s*invc;
    float var = s2*invc - mu*mu;
    float sc = rsqrtf(var + GN_EPS) * w[ch];
    cf[i*2]   = sc;
    cf[i*2+1] = bb[ch] - mu*sc;
  }
}

// final GN apply (no relu)
__global__ __launch_bounds__(256) void k_apply_coef(const float* __restrict__ x,
                                                    float* __restrict__ y,
                                                    const float* __restrict__ cf,
                                                    int C, int spatial, size_t total) {
  const size_t per = (size_t)C * spatial;
  for (size_t i = (size_t)blockIdx.x*blockDim.x + threadIdx.x; i < total;
       i += (size_t)gridDim.x*blockDim.x) {
    int b = (int)(i / per);
    int ch = (int)((i - (size_t)b*per) / spatial);
    y[i] = x[i]*cf[(b*C+ch)*2] + cf[(b*C+ch)*2+1];
  }
}

// fused: y = relu(GN2(out2)) + relu(GNsc(resid)), + GN_out partial stats.
// grid = (64, 4, B)
__global__ __launch_bounds__(256) void k_fuse_add(const float* __restrict__ x2,
                                                  const float* __restrict__ xr,
                                                  const float* __restrict__ cf2,
                                                  const float* __restrict__ cfS,
                                                  float* __restrict__ y,
                                                  float* __restrict__ part) {
  const int b = blockIdx.z, grp = blockIdx.y, p = blockIdx.x;
  const int nparts = gridDim.x, tid = threadIdx.x;
  const size_t cnt = (size_t)(COUT/4) * NPER;
  const size_t base = ((size_t)b*4 + grp) * cnt;
  float s = 0.f, s2 = 0.f;
  for (size_t i = (size_t)p*256 + tid; i < cnt; i += (size_t)nparts*256) {
    int ch = grp*(COUT/4) + (int)(i / NPER);
    size_t idx = base + i;
    float a = fmaxf(x2[idx]*cf2[(b*COUT+ch)*2] + cf2[(b*COUT+ch)*2+1], 0.f);
    float r = fmaxf(xr[idx]*cfS[(b*COUT+ch)*2] + cfS[(b*COUT+ch)*2+1], 0.f);
    float v = a + r;
    y[idx] = v;
    s += v; s2 += v*v;
  }
  __shared__ float r1[256], r2[256];
  r1[tid] = s; r2[tid] = s2;
  __syncthreads();
  for (int o = 128; o > 0; o >>= 1) {
    if (tid < o) { r1[tid] += r1[tid+o]; r2[tid] += r2[tid+o]; }
    __syncthreads();
  }
  if (tid == 0) {
    int bg = b*4 + grp;
    part[((size_t)bg*nparts + p)*2]     = r1[0];
    part[((size_t)bg*nparts + p)*2 + 1] = r2[0];
  }
}

// ---------------- fused masked attention ----------------
static __device__ __forceinline__ void attn_kv_tile(const float* __restrict__ corr,
                                                    const float* __restrict__ Wrow,
                                                    const float* __restrict__ brow,
                                                    float (*dst)[68],
                                                    int b, int q, int s0,
                                                    int lane, int wv) {
  const int et = wv >> 2, st = wv & 3;
  const int er  = lane & 15;
  const int kof = (lane < 16) ? 0 : 2;
  const int scol = st*16 + (lane & 15);
  const int sg = min(s0 + scol, TS - 1);
  const float* wr = Wrow + (size_t)(et*16 + er) * CIN;
  const size_t cstr = (size_t)TQ * TS;
  const float* cp = corr + (size_t)b * CIN * cstr + (size_t)q * TS + sg;
  v8f acc = zero8();
  for (int kk = 0; kk < CIN/4; ++kk) {
    int c0 = kk*4 + kof;
    v2f a;  a.x = wr[c0];               a.y = wr[c0 + 1];
    v2f bv; bv.x = cp[(size_t)c0*cstr]; bv.y = cp[(size_t)(c0+1)*cstr];
    acc = wmma4(a, bv, acc);
  }
  const int rb = (lane < 16) ? 0 : 8;
#pragma unroll
  for (int r = 0; r < 8; ++r) {
    int e = et*16 + rb + r;
    dst[e][scol] = acc[r] + brow[e];
  }
}

static __device__ __forceinline__ void attn_score(const float (*qs)[80],
                                                  const float (*kb)[68],
                                                  float (*ab)[68],
                                                  const float* __restrict__ mk,
                                                  int s0, int lane, int wv) {
  for (int t = wv; t < 20; t += 8) {
    const int dt = t >> 2, st = t & 3;
    const int drow = dt*16 + (lane & 15);
    const int eof  = (lane < 16) ? 0 : 2;
    const int scol = st*16 + (lane & 15);
    v8f acc = zero8();
    for (int kk = 0; kk < 8; ++kk) {
      int e0 = kk*4 + eof;
      v2f a;  a.x = qs[e0][drow];  a.y = qs[e0+1][drow];
      v2f bm; bm.x = kb[e0][scol]; bm.y = kb[e0+1][scol];
      acc = wmma4(a, bm, acc);
    }
    const int sg = s0 + scol;
    const bool valid = (sg < TS) && (sg == 0 || mk[sg-1] != 0.f);
    const int rb = (lane < 16) ? 0 : 8;
#pragma unroll
    for (int r = 0; r < 8; ++r)
      ab[dt*16 + rb + r][scol] = valid ? acc[r] : -1.0e9f;
  }
}

static __device__ __forceinline__ void pv_accum(const float (*vbuf)[68],
                                                const float (*abuf)[68],
                                                int t, int lane, v8f& acc) {
  const int et = t / 5, dt = t % 5;
  const int er  = lane & 15;
  const int sof = (lane < 16) ? 0 : 2;
  const int dcol = dt*16 + (lane & 15);
  for (int kk = 0; kk < 16; ++kk) {
    int ss = kk*4 + sof;
    v2f a;  a.x = vbuf[et*16 + er][ss]; a.y = vbuf[et*16 + er][ss+1];
    v2f bm; bm.x = abuf[dcol][ss];      bm.y = abuf[dcol][ss+1];
    acc = wmma4(a, bm, acc);
  }
}

static __device__ __forceinline__ void attn_write(float* __restrict__ out,
                                                  const float* lsum, v8f acc,
                                                  int t, int lane,
                                                  int b, int g, int q) {
  const int et = t / 5, dt = t % 5;
  const int dcol = dt*16 + (lane & 15);
  if (dcol >= TD) return;
  const float ls = lsum[dcol];
  const int rb = (lane < 16) ? 0 : 8;
#pragma unroll
  for (int r = 0; r < 8; ++r) {
    int e = et*16 + rb + r;
    out[(((size_t)b*HID + g*EDIM + e)*TQ + q)*TD + dcol] = acc[r] / ls;
  }
}

// issue one 2D TDM tile load: rows x tileW elems (f32), row stride in elems,
// OOB columns (>= tensW) fill LDS with zeros.
static __device__ __forceinline__ void tdm_load_2d(const float* gptr,
                                                   uint32_t lds_addr,
                                                   int tensW, int rows,
                                                   int tileW, int rstride) {
  uint64_t ga = (uint64_t)(uintptr_t)gptr;
  u32x4 g0;
  g0.x = 1u;                                   // count=1, user descriptor
  g0.y = lds_addr;                             // LDS byte address
  g0.z = (uint32_t)ga;                         // global_addr[31:0]
  g0.w = (uint32_t)((ga >> 32) & 0x01FFFFFFu) | (2u << 30);  // addr[56:32], type=2
  i32x8 g1;
  g1[0] = (int)(2u << 16);                     // data_size = 4B
  g1[1] = (int)((uint32_t)tensW << 16);        // tensor_dim0[15:0]
  g1[2] = (int)((uint32_t)rows  << 16);        // tensor_dim0[31:16]=0, tensor_dim1[15:0]
  g1[3] = (int)((uint32_t)tileW << 16);        // tensor_dim1[31:16]=0, tile_dim0
  g1[4] = rows;                                // tile_dim1, tile_dim2=0
  g1[5] = rstride;                             // tensor_dim0_stride[31:0]
  g1[6] = 0; g1[7] = 0;
  i32x4 gz; gz[0] = 0; gz[1] = 0; gz[2] = 0; gz[3] = 0;
  i32x8 gz8;
  gz8[0] = 0; gz8[1] = 0; gz8[2] = 0; gz8[3] = 0;
  gz8[4] = 0; gz8[5] = 0; gz8[6] = 0; gz8[7] = 0;
  __builtin_amdgcn_tensor_load_to_lds(g0, g1, gz, gz, gz8, 0);
}

// grid = (NHEAD, TQ, B), 256 threads (8 waves)
__global__ __launch_bounds__(256) void k_attn(const float* __restrict__ corr,
                                              const float* __restrict__ Wqkv,
                                              const float* __restrict__ bqkv,
                                              const float* __restrict__ Q,
                                              const float* __restrict__ maskb,
                                              float* __restrict__ out,
                                              float* __restrict__ part) {
  __shared__ float qs[EDIM][80];   // q[e][d]; cols 65..79 zero-filled by TDM OOB
  __shared__ float kb[EDIM][68];
  __shared__ float vb[EDIM][68];
  __shared__ float ab[80][68];
  __shared__ float mk[256];
  __shared__ float mmax[80];
  __shared__ float lsum[80];

  const int g = blockIdx.x, q = blockIdx.y, b = blockIdx.z;
  const int tid = threadIdx.x, lane = tid & 31, wv = tid >> 5;

  const float* Wk  = Wqkv + (size_t)(HID   + g*EDIM) * CIN;
  const float* Wv  = Wqkv + (size_t)(2*HID + g*EDIM) * CIN;
  const float* bk  = bqkv + HID   + g*EDIM;
  const float* bvv = bqkv + 2*HID + g*EDIM;

  // Tensor Data Mover: stage Q tile (32 x 65 -> padded 32 x 80) and mask row
  if (wv == 0) {
    tdm_load_2d(Q + (((size_t)b*HID + g*EDIM)*TQ + q)*TD,
                (uint32_t)(uintptr_t)(&qs[0][0]),
                TD, EDIM, 80, TQ*TD);
    tdm_load_2d(maskb + (size_t)b*256,
                (uint32_t)(uintptr_t)(&mk[0]),
                256, 1, 256, 256);
    __builtin_amdgcn_s_wait_tensorcnt(0);
  }
  if (tid < 80) { mmax[tid] = -3.0e38f; lsum[tid] = 0.f; }
  __syncthreads();

  // ---- pass A: row maxima over all s ----
  for (int ch = 0; ch < 5; ++ch) {
    const int s0 = ch * 64;
    if (ch < 4) {   // prefetch next s-chunk of the corr slice
      int c = tid & 127;
      int sp = s0 + 64 + ((tid >> 7) << 5);
      if (sp < TS)
        __builtin_prefetch(corr + ((size_t)b*CIN + c)*((size_t)TQ*TS)
                                + (size_t)q*TS + sp, 0, 1);
    }
    attn_kv_tile(corr, Wk, bk, kb, b, q, s0, lane, wv);
    __syncthreads();
    attn_score(qs, kb, ab, mk, s0, lane, wv);
    __syncthreads();
    if (tid < 80) {
      float m_ = mmax[tid];
      for (int j = 0; j < 64; ++j) m_ = fmaxf(m_, ab[tid][j]);
      mmax[tid] = m_;
    }
    __syncthreads();
  }

  // ---- pass B: exp/sum + PV with persistent accumulators ----
  v8f o0 = zero8(), o1 = zero8();
  for (int ch = 0; ch < 5; ++ch) {
    const int s0 = ch * 64;
    attn_kv_tile(corr, Wk, bk,  kb, b, q, s0, lane, wv);
    attn_kv_tile(corr, Wv, bvv, vb, b, q, s0, lane, wv);
    __syncthreads();
    attn_score(qs, kb, ab, mk, s0, lane, wv);
    __syncthreads();
    if (tid < 80) {
      float m_ = mmax[tid], rs = 0.f;
      for (int j = 0; j < 64; ++j) {
        float e_ = __expf(ab[tid][j] - m_);
        ab[tid][j] = e_;
        rs += e_;
      }
      lsum[tid] += rs;
    }
    __syncthreads();
    pv_accum(vb, ab, wv, lane, o0);
    if (wv < 2) pv_accum(vb, ab, wv + 8, lane, o1);
    __syncthreads();
  }

  attn_write(out, lsum, o0, wv, lane, b, g, q);
  if (wv < 2) attn_write(out, lsum, o1, wv + 8, lane, b, g, q);

  // ---- GN1 partial stats for this block's 32x65 output tile ----
  float s = 0.f, s2 = 0.f;
  {
    int dt = wv % 5;
    int dcol = dt*16 + (lane & 15);
    if (dcol < TD) {
      float ls = lsum[dcol];
#pragma unroll
      for (int r = 0; r < 8; ++r) { float v = o0[r]/ls; s += v; s2 += v*v; }
    }
    if (wv < 2) {
      int t1 = wv + 8;
      dt = t1 % 5;
      dcol = dt*16 + (lane & 15);
      if (dcol < TD) {
        float ls = lsum[dcol];
#pragma unroll
        for (int r = 0; r < 8; ++r) { float v = o1[r]/ls; s += v; s2 += v*v; }
      }
    }
  }
  float* r1 = &ab[0][0];       // reuse LDS (last read was before loop-end barrier)
  float* r2 = r1 + 256;
  r1[tid] = s; r2[tid] = s2;
  __syncthreads();
  for (int o = 128; o > 0; o >>= 1) {
    if (tid < o) { r1[tid] += r1[tid+o]; r2[tid] += r2[tid+o]; }
    __syncthreads();
  }
  if (tid == 0) {
    int bg = b*4 + (g >> 1);                 // chPerGrp=64 => 2 heads per group
    size_t pidx = (size_t)bg*512 + (g & 1)*256 + q;
    part[pidx*2]     = r1[0];
    part[pidx*2 + 1] = r2[0];
  }
}

// ---------------- host-side launch ----------------
extern "C" void kernel_launch(void* const* d_in, const int* in_sizes, int n_in,
                              void* d_out, int out_size, void* d_ws, size_t ws_size,
                              hipStream_t stream) {
  const float* corr    = (const float*)d_in[0];
  const int*   smask   = (const int*)  d_in[1];
  const float* W_sc    = (const float*)d_in[2];
  const float* gn_sc_w = (const float*)d_in[3];
  const float* gn_sc_b = (const float*)d_in[4];
  const float* W_qkv   = (const float*)d_in[5];
  const float* b_qkv   = (const float*)d_in[6];
  const float* gn1_w   = (const float*)d_in[7];
  const float* gn1_b   = (const float*)d_in[8];
  const float* W_agg   = (const float*)d_in[9];
  const float* gn2_w   = (const float*)d_in[10];
  const float* gn2_b   = (const float*)d_in[11];
  const float* gno_w   = (const float*)d_in[12];
  const float* gno_b   = (const float*)d_in[13];
  float* out = (float*)d_out;

  // workspace layout (floats)
  float* ws = (float*)d_ws;
  size_t o = 0;
  float* corrP = ws + o; o += (size_t)B_ * CIN  * NPER;
  float* Qbuf  = ws + o; o += (size_t)B_ * HID  * NPER;
  float* resid = ws + o; o += (size_t)B_ * COUT * NPER;
  float* attno = ws + o; o += (size_t)B_ * HID  * NPER;
  float* out2  = ws + o; o += (size_t)B_ * COUT * NPER;
  float* maskb = ws + o; o += (size_t)B_ * 256;
  float* partA = ws + o; o += (size_t)16 * 512 * 2;   // max nparts = 512
  float* stats = ws + o; o += 32;
  float* coefS = ws + o; o += (size_t)B_ * COUT * 2;
  float* coef1 = ws + o; o += (size_t)B_ * HID  * 2;
  float* coef2 = ws + o; o += (size_t)B_ * COUT * 2;
  float* coefO = ws + o; o += (size_t)B_ * COUT * 2;
  (void)o; (void)ws_size; (void)in_sizes; (void)n_in; (void)out_size;

  const size_t totC = (size_t)B_ * COUT * NPER;
  const float invC512 = 1.0f / (128.0f * (float)NPER);
  const float invC256 = 1.0f / ( 64.0f * (float)NPER);

  k_mask<<<B_, 256, 0, stream>>>(smask, maskb);
  k_pool<<<16384, 256, 0, stream>>>(corr, corrP, (size_t)B_ * CIN * NPER);

  // residual conv (raw) + fused GN_sc stats
  k_gemm_t<false><<<dim3(NPER/64, COUT/256, B_), 256, 0, stream>>>(
      W_sc, corrP, nullptr, nullptr, resid, partA, COUT, CIN);
  k_gn_reduceN<<<16, 256, 0, stream>>>(partA, stats, NPER/64);
  k_gn_coef<<<8, 256, 0, stream>>>(stats, gn_sc_w, gn_sc_b, coefS, COUT, invC512);

  // Q projection on pooled input
  k_gemm_t<false><<<dim3(NPER/64, HID/256, B_), 256, 0, stream>>>(
      W_qkv, corrP, b_qkv, nullptr, Qbuf, nullptr, HID, CIN);

  // fused attention (TDM-staged Q/mask, K/V recomputed via WMMA) + GN1 stats
  k_attn<<<dim3(NHEAD, TQ, B_), 256, 0, stream>>>(
      corr, W_qkv, b_qkv, Qbuf, maskb, attno, partA);
  k_gn_reduceN<<<16, 256, 0, stream>>>(partA, stats, 512);
  k_gn_coef<<<8, 256, 0, stream>>>(stats, gn1_w, gn1_b, coef1, HID, invC256);

  // aggregation conv with fused GN1-apply+relu on input, + GN2 stats
  k_gemm_t<true><<<dim3(NPER/64, COUT/256, B_), 256, 0, stream>>>(
      W_agg, attno, nullptr, coef1, out2, partA, COUT, HID);
  k_gn_reduceN<<<16, 256, 0, stream>>>(partA, stats, NPER/64);
  k_gn_coef<<<8, 256, 0, stream>>>(stats, gn2_w, gn2_b, coef2, COUT, invC512);

  // relu(GN2(out2)) + relu(GNsc(resid)) -> d_out, + GN_out stats
  k_fuse_add<<<dim3(64, 4, B_), 256, 0, stream>>>(
      out2, resid, coef2, coefS, out, partA);
  k_gn_reduceN<<<16, 256, 0, stream>>>(partA, stats, 64);
  k_gn_coef<<<8, 256, 0, stream>>>(stats, gno_w, gno_b, coefO, COUT, invC512);

  // final GN apply in-place
  k_apply_coef<<<4096, 256, 0, stream>>>(out, out, coefO, COUT, NPER, totC);
}